// SymmetricContractionWrapper_21174188770060
// MI455X (gfx1250) — compile-verified
//
#include <hip/hip_runtime.h>

// MACE symmetric contraction collapsed to a single bilinear form per (b,c):
//   final[b,c] = xx^T * [U3|U2] * zvec + (U1 term),  xx = x (x) x
// K folds 256 -> 136 via xx symmetry (B' = U3[p,q]+U3[q,p] off-diagonal);
// fp32 WMMA (v_wmma_f32_16x16x4_f32) keeps full fidelity with the reference.
// N (192) in two sequential passes of 6 n-tiles keeps the live set ~150 VGPRs;
// amdgpu_num_vgpr(256) pins the allocation below the MSB boundary so the hot
// loop has no s_set_vgpr_msb churn and 4 waves/SIMD can co-reside with the
// two LDS-resident workgroups per 320 KiB WGP.

typedef __attribute__((ext_vector_type(2))) float v2f;
typedef __attribute__((ext_vector_type(8))) float v8f;

namespace {
constexpr int Bn = 2048, Cn = 128, In = 16, En = 10;
constexpr int K3 = 11, K2 = 4;
constexpr int KK = 136;          // folded GEMM K: 16 diag + 120 upper-tri (p,q)
constexpr int NSTEP = KK / 4;    // 34 wmma k-steps (no K padding)
constexpr int ND = 192;          // GEMM N: 176 (i,k3) + 4 (k2) + 12 zero pad
constexpr int NTH = 6;           // n-tiles per pass
constexpr int NPASS = 2;         // 2 passes x 6 tiles = 12 n-tiles
constexpr int BPB = 4;           // b-values per block (amortize U3 -> LDS staging)

// LDS layout (float offsets)
constexpr int OFF_BT = 0;                    // [ND][KK]  folded B^T : BT[n][kk]
constexpr int OFF_X  = OFF_BT + ND * KK;     // [Cn][In]
constexpr int OFF_W3 = OFF_X  + Cn * In;     // [K3][Cn]
constexpr int OFF_W2 = OFF_W3 + K3 * Cn;     // [K2][Cn]
constexpr int OFF_T1 = OFF_W2 + K2 * Cn;     // [Cn]
constexpr int OFF_PQ = OFF_T1 + Cn;          // [KK] packed (p | q<<8) ints
constexpr int LDS_FLOATS = OFF_PQ + KK;      // 30344 floats ~ 118.5 KiB -> 2 blocks/WGP
constexpr int LDS_BYTES  = LDS_FLOATS * 4;
}

__global__ void __launch_bounds__(256) __attribute__((amdgpu_num_vgpr(256)))
mace_symmetric_contraction(const float* __restrict__ x,
                           const float* __restrict__ y,
                           const float* __restrict__ U3,
                           const float* __restrict__ U2,
                           const float* __restrict__ U1,
                           const float* __restrict__ W3,
                           const float* __restrict__ W2,
                           const float* __restrict__ W1,
                           float* __restrict__ out)
{
    extern __shared__ float lds[];
    const int tid  = threadIdx.x;
    const int lane = tid & 31;
    const int wave = tid >> 5;           // 8 waves; wave w owns channels [16w,16w+16)
    const int hlf  = lane >> 4;          // K/M half select per WMMA VGPR layout
    const int col  = lane & 15;          // M for A, N for B/D
    const int c0   = wave * 16;

    int* ipq = (int*)&lds[OFF_PQ];

    // ---- packed (p,q) table for the folded K index: kk<16 -> diag, else p<q pairs
    if (tid < KK) {
        int p, q;
        if (tid < 16) { p = q = tid; }
        else {
            int m = tid - 16, c = 15;
            p = 0;
            while (m >= c) { m -= c; ++p; --c; }
            q = p + 1 + m;
        }
        ipq[tid] = p | (q << 8);
    }
    __syncthreads();

    // ---- stage folded B^T = [U3sym | U2sym | 0] as BT[n][kk], once per block
    for (int u = tid; u < ND * KK; u += 256) {
        const int kk = u % KK;
        const int n  = u / KK;
        const int pqv = ipq[kk];
        const int p = pqv & 255, q = pqv >> 8;
        float v;
        if (n < 176) {
            const int i  = n / K3;
            const int k3 = n - i * K3;
            v = U3[((p * 16 + q) * 16 + i) * K3 + k3];
            if (p != q) v += U3[((q * 16 + p) * 16 + i) * K3 + k3];
        } else if (n < 180) {
            const int k2 = n - 176;
            v = U2[(p * 16 + q) * K2 + k2];
            if (p != q) v += U2[(q * 16 + p) * K2 + k2];
        } else {
            v = 0.f;
        }
        lds[OFF_BT + n * KK + kk] = v;
    }

    // per-lane operand pointers (8B-aligned: col*KK and 2*hlf both even)
    const v2f*   btv  = (const v2f*)&lds[OFF_BT + col * KK + 2 * hlf];
    const float* xrow = &lds[OFF_X + (c0 + col) * In];

    for (int rb = 0; rb < BPB; ++rb) {
        const int b = blockIdx.x * BPB + rb;

        // species index: argmax(y[b,:]), first-occurrence tie-break
        int e = 0;
        float best = y[b * En];
        for (int t = 1; t < En; ++t) {
            float v = y[b * En + t];
            if (v > best) { best = v; e = t; }
        }

        __syncthreads();  // previous iteration's readers of sX / W tiles are done
        for (int u = tid; u < Cn * In; u += 256) lds[OFF_X  + u] = x[(size_t)b * Cn * In + u];
        for (int u = tid; u < K3 * Cn; u += 256) lds[OFF_W3 + u] = W3[e * K3 * Cn + u];
        for (int u = tid; u < K2 * Cn; u += 256) lds[OFF_W2 + u] = W2[e * K2 * Cn + u];
        __syncthreads();
        if (tid < Cn) {  // correlation-1 term (K1==1)
            float s = 0.f;
            #pragma unroll
            for (int p = 0; p < In; ++p) s += lds[OFF_X + tid * In + p] * U1[p];
            lds[OFF_T1 + tid] = s * W1[e * Cn + tid];
        }
        __syncthreads();

        auto loadA = [&](int s) -> v2f {
            const int base = 4 * s + 2 * hlf;
            const int pq0 = ipq[base];
            const int pq1 = ipq[base + 1];
            v2f A;
            A.x = xrow[pq0 & 255] * xrow[pq0 >> 8];
            A.y = xrow[pq1 & 255] * xrow[pq1 >> 8];
            return A;
        };

        float res[8];
        #pragma unroll
        for (int j = 0; j < 8; ++j) res[j] = 0.f;

        // ---- two sequential N passes: 6 n-tiles each, full folded K per pass
        for (int pass = 0; pass < NPASS; ++pass) {      // runtime loop: passes not interleaved
            const v2f* btp = btv + (size_t)pass * NTH * (8 * KK);

            v8f acc[NTH] = {};
            v2f A0, B0[NTH];

            #pragma unroll
            for (int t = 0; t < NTH; ++t) B0[t] = btp[t * (8 * KK)];
            A0 = loadA(0);

            for (int s = 0; s < NSTEP - 1; ++s) {
                v2f An = loadA(s + 1);
                #pragma unroll
                for (int t = 0; t < NTH; ++t) {
                    acc[t] = __builtin_amdgcn_wmma_f32_16x16x4_f32(
                        false, A0, false, B0[t], (short)0, acc[t], false, false);
                    B0[t] = btp[t * (8 * KK) + 2 * (s + 1)];   // prefetch next k-step
                }
                A0 = An;
            }
            #pragma unroll
            for (int t = 0; t < NTH; ++t)
                acc[t] = __builtin_amdgcn_wmma_f32_16x16x4_f32(
                    false, A0, false, B0[t], (short)0, acc[t], false, false);

            // epilogue half: res[j] += sum_n Y[m][n]*z[m][n],  z = x_i*w3_k3 | w2_k2 | 0
            #pragma unroll
            for (int tt = 0; tt < NTH; ++tt) {
                const int n  = (pass * NTH + tt) * 16 + col;
                const int i  = n / K3;
                const int k3 = n - i * K3;
                #pragma unroll
                for (int j = 0; j < 8; ++j) {
                    const int cm = c0 + j + 8 * hlf;   // D layout: row = j + 8*half
                    float zz;
                    if (n < 176)       zz = lds[OFF_X + cm * In + i] * lds[OFF_W3 + k3 * Cn + cm];
                    else if (n < 180)  zz = lds[OFF_W2 + (n - 176) * Cn + cm];
                    else               zz = 0.f;
                    res[j] += acc[tt][j] * zz;
                }
            }
        }

        // reduce the 16 column lanes of each half, add T1, write out[b,c]
        #pragma unroll
        for (int j = 0; j < 8; ++j) {
            float v = res[j];
            v += __shfl_xor(v, 1, 32);
            v += __shfl_xor(v, 2, 32);
            v += __shfl_xor(v, 4, 32);
            v += __shfl_xor(v, 8, 32);
            if (col == j) {
                const int m = j + 8 * hlf;
                out[(size_t)b * Cn + c0 + m] = v + lds[OFF_T1 + c0 + m];
            }
        }
    }
}

extern "C" void kernel_launch(void* const* d_in, const int* in_sizes, int n_in,
                              void* d_out, int out_size, void* d_ws, size_t ws_size,
                              hipStream_t stream) {
    (void)in_sizes; (void)n_in; (void)out_size; (void)d_ws; (void)ws_size;
    const float* x  = (const float*)d_in[0];
    const float* y  = (const float*)d_in[1];
    const float* U3 = (const float*)d_in[2];
    const float* U2 = (const float*)d_in[3];
    const float* U1 = (const float*)d_in[4];
    const float* W3 = (const float*)d_in[5];
    const float* W2 = (const float*)d_in[6];
    const float* W1 = (const float*)d_in[7];
    float* out = (float*)d_out;

    // ~119 KiB dynamic LDS: 2 workgroups per 320 KiB CDNA5 WGP
    (void)hipFuncSetAttribute((const void*)mace_symmetric_contraction,
                              hipFuncAttributeMaxDynamicSharedMemorySize, LDS_BYTES);

    dim3 grid(Bn / BPB), block(256);
    mace_symmetric_contraction<<<grid, block, LDS_BYTES, stream>>>(
        x, y, U3, U2, U1, W3, W2, W1, out);
}